// GCN_59622736003658
// MI455X (gfx1250) — compile-verified
//
#include <hip/hip_runtime.h>
#include <math.h>

// ---------------- constants from the reference ----------------
#define N_NODES 50000
#define N_EDGES 1600000
#define NFEAT   256
#define NHID    128
#define NCLASS  40

typedef float v2f __attribute__((ext_vector_type(2)));
typedef float v8f __attribute__((ext_vector_type(8)));

// ---------------- utility fills ----------------
__global__ void gcn_fill_f32(float* __restrict__ p, float v, int n) {
    int i = blockIdx.x * blockDim.x + threadIdx.x;
    if (i < n) p[i] = v;
}

// ---------------- degree / normalization ----------------
__global__ void gcn_deg_accum(const int* __restrict__ dst, float* __restrict__ deg, int E) {
    int e = blockIdx.x * blockDim.x + threadIdx.x;
    if (e < E) atomicAdd(&deg[dst[e]], 1.0f);
}

__global__ void gcn_deg_to_dinv(float* __restrict__ deg, int n) {
    int i = blockIdx.x * blockDim.x + threadIdx.x;
    if (i < n) deg[i] = rsqrtf(fmaxf(deg[i], 1.0f));
}

// ---------------- GEMM1: H[N x 128] = X[N x 256] @ W1[256 x 128], fp32 WMMA ----------------
// One wave per 16-row strip; the wave produces all 8 column tiles (16x128),
// so each X element is loaded exactly once per wave.
__global__ void gcn_gemm1_wmma(const float* __restrict__ X, const float* __restrict__ W,
                               float* __restrict__ H) {
    int wave = (blockIdx.x * blockDim.x + threadIdx.x) >> 5;
    int lane = threadIdx.x & 31;
    if (wave >= N_NODES / 16) return;          // whole-wave uniform exit, EXEC stays all-1s

    const int row0 = wave * 16;
    const int m  = lane & 15;                  // A row / B,C column within tile
    const int kb = (lane >> 4) * 2;            // lane-group K base (0 or 2)

    v8f acc[8] = {};
    const float* xrow = X + (size_t)(row0 + m) * NFEAT + kb;

    for (int k = 0; k < NFEAT; k += 4) {
        v2f a;
        a.x = xrow[k];                         // K = kb
        a.y = xrow[k + 1];                     // K = kb+1
        #pragma unroll
        for (int t = 0; t < 8; ++t) {
            int n = t * 16 + m;
            v2f b;
            b.x = W[(size_t)(k + kb) * NHID + n];
            b.y = W[(size_t)(k + kb + 1) * NHID + n];
            acc[t] = __builtin_amdgcn_wmma_f32_16x16x4_f32(
                false, a, false, b, (short)0, acc[t], false, false);
        }
    }

    const int srow = row0 + 8 * (lane >> 4);   // C/D: VGPR r -> M = r + 8*(lane>=16)
    #pragma unroll
    for (int t = 0; t < 8; ++t) {
        int col = t * 16 + m;
        #pragma unroll
        for (int r = 0; r < 8; ++r)
            H[(size_t)(srow + r) * NHID + col] = acc[t][r];
    }
}

// ---------------- GEMM2: H2[N x 40] = Hin[N x 128] @ W2[128 x 40], fp32 WMMA ----------------
__global__ void gcn_gemm2_wmma(const float* __restrict__ A, const float* __restrict__ W,
                               float* __restrict__ H) {
    int wave = (blockIdx.x * blockDim.x + threadIdx.x) >> 5;
    int lane = threadIdx.x & 31;
    if (wave >= N_NODES / 16) return;

    const int row0 = wave * 16;
    const int m  = lane & 15;
    const int kb = (lane >> 4) * 2;

    v8f acc[3] = {};
    const float* arow = A + (size_t)(row0 + m) * NHID + kb;

    for (int k = 0; k < NHID; k += 4) {
        v2f a;
        a.x = arow[k];
        a.y = arow[k + 1];
        #pragma unroll
        for (int t = 0; t < 3; ++t) {
            int n = t * 16 + m;
            v2f b;
            if (n < NCLASS) {
                b.x = W[(size_t)(k + kb) * NCLASS + n];
                b.y = W[(size_t)(k + kb + 1) * NCLASS + n];
            } else {
                b.x = 0.0f; b.y = 0.0f;
            }
            acc[t] = __builtin_amdgcn_wmma_f32_16x16x4_f32(
                false, a, false, b, (short)0, acc[t], false, false);
        }
    }

    const int srow = row0 + 8 * (lane >> 4);
    #pragma unroll
    for (int t = 0; t < 3; ++t) {
        int col = t * 16 + m;
        if (col < NCLASS) {
            #pragma unroll
            for (int r = 0; r < 8; ++r)
                H[(size_t)(srow + r) * NCLASS + col] = acc[t][r];
        }
    }
}

// ---------------- edge aggregation: one wave per edge, float4 gather, fp32 atomic scatter ----
__global__ void gcn_edge_agg(const int* __restrict__ src, const int* __restrict__ dst,
                             const float* __restrict__ dinv, const float* __restrict__ Hin,
                             float* __restrict__ Hout, int E, int F) {
    int e    = (blockIdx.x * blockDim.x + threadIdx.x) >> 5;
    int lane = threadIdx.x & 31;
    if (e >= E) return;
    int s = src[e], d = dst[e];
    float norm = dinv[s] * dinv[d];
    int nvec = F >> 2;
    const float4* in = (const float4*)(Hin + (size_t)s * F);
    float* outrow = Hout + (size_t)d * F;
    for (int v = lane; v < nvec; v += 32) {
        float4 val = in[v];
        atomicAdd(outrow + 4 * v + 0, val.x * norm);
        atomicAdd(outrow + 4 * v + 1, val.y * norm);
        atomicAdd(outrow + 4 * v + 2, val.z * norm);
        atomicAdd(outrow + 4 * v + 3, val.w * norm);
    }
}

// ---------------- layer-1 epilogue: self-loop + bias + ReLU + dropout mask (in place) -------
__global__ void gcn_post1(float* __restrict__ Agg, const float* __restrict__ H1,
                          const float* __restrict__ dinv, const float* __restrict__ b1,
                          const float* __restrict__ mask) {
    int idx = blockIdx.x * blockDim.x + threadIdx.x;
    if (idx >= N_NODES * NHID) return;
    int i = idx >> 7;           // / NHID
    int f = idx & (NHID - 1);
    float di = dinv[i];
    float v = Agg[idx] + H1[idx] * di * di + b1[f];
    v = fmaxf(v, 0.0f) * mask[idx];
    Agg[idx] = v;
}

// ---------------- layer-2 epilogue: self-loop + bias + log_softmax (one wave per node) ------
__global__ void gcn_post2_logsoftmax(float* __restrict__ out, const float* __restrict__ H2,
                                     const float* __restrict__ dinv, const float* __restrict__ b2) {
    int node = (blockIdx.x * blockDim.x + threadIdx.x) >> 5;
    int lane = threadIdx.x & 31;
    if (node >= N_NODES) return;
    float di = dinv[node];
    float di2 = di * di;
    const size_t base = (size_t)node * NCLASS;

    float x0 = out[base + lane] + H2[base + lane] * di2 + b2[lane];
    float x1 = -3.0e38f;
    if (lane < NCLASS - 32) {
        int f = lane + 32;
        x1 = out[base + f] + H2[base + f] * di2 + b2[f];
    }

    float mx = fmaxf(x0, x1);
    for (int off = 16; off; off >>= 1) mx = fmaxf(mx, __shfl_xor(mx, off, 32));

    float e = expf(x0 - mx) + ((lane < NCLASS - 32) ? expf(x1 - mx) : 0.0f);
    for (int off = 16; off; off >>= 1) e += __shfl_xor(e, off, 32);
    float lse = logf(e);

    out[base + lane] = x0 - mx - lse;
    if (lane < NCLASS - 32) out[base + lane + 32] = x1 - mx - lse;
}

// ---------------- host side ----------------
extern "C" void kernel_launch(void* const* d_in, const int* in_sizes, int n_in,
                              void* d_out, int out_size, void* d_ws, size_t ws_size,
                              hipStream_t stream) {
    (void)in_sizes; (void)n_in; (void)out_size; (void)ws_size;

    const float* x    = (const float*)d_in[0];
    const int*   eidx = (const int*)  d_in[1];   // [2, E] int32
    const float* W1   = (const float*)d_in[2];
    const float* b1   = (const float*)d_in[3];
    const float* W2   = (const float*)d_in[4];
    const float* b2   = (const float*)d_in[5];
    const float* mask = (const float*)d_in[6];
    float* out = (float*)d_out;

    const int* src = eidx;
    const int* dst = eidx + N_EDGES;

    // workspace layout (floats): dinv[N] | bufH[N*128] | bufAgg[N*128]
    float* dinv   = (float*)d_ws;
    float* bufH   = dinv + N_NODES;                       // H1, later reused as H2 (N x 40)
    float* bufAgg = bufH + (size_t)N_NODES * NHID;

    const int BLK = 256;

    // 1. init: deg=1 (self loop), bufAgg=0, out=0
    gcn_fill_f32<<<(N_NODES + BLK - 1) / BLK, BLK, 0, stream>>>(dinv, 1.0f, N_NODES);
    gcn_fill_f32<<<(N_NODES * NHID + BLK - 1) / BLK, BLK, 0, stream>>>(bufAgg, 0.0f, N_NODES * NHID);
    gcn_fill_f32<<<(N_NODES * NCLASS + BLK - 1) / BLK, BLK, 0, stream>>>(out, 0.0f, N_NODES * NCLASS);

    // 2. degrees -> D^{-1/2}
    gcn_deg_accum<<<(N_EDGES + BLK - 1) / BLK, BLK, 0, stream>>>(dst, dinv, N_EDGES);
    gcn_deg_to_dinv<<<(N_NODES + BLK - 1) / BLK, BLK, 0, stream>>>(dinv, N_NODES);

    // 3. H1 = X @ W1   (WMMA fp32)
    {
        int waves = N_NODES / 16;                          // 3125
        int blocks = (waves * 32 + BLK - 1) / BLK;
        gcn_gemm1_wmma<<<blocks, BLK, 0, stream>>>(x, W1, bufH);
    }

    // 4. edge aggregation layer 1
    {
        long long threads = (long long)N_EDGES * 32;
        int blocks = (int)((threads + BLK - 1) / BLK);
        gcn_edge_agg<<<blocks, BLK, 0, stream>>>(src, dst, dinv, bufH, bufAgg, N_EDGES, NHID);
    }

    // 5. self-loop + bias + relu + dropout
    gcn_post1<<<(N_NODES * NHID + BLK - 1) / BLK, BLK, 0, stream>>>(bufAgg, bufH, dinv, b1, mask);

    // 6. H2 = H1' @ W2  (WMMA fp32), reuse bufH as N x 40
    {
        int waves = N_NODES / 16;
        int blocks = (waves * 32 + BLK - 1) / BLK;
        gcn_gemm2_wmma<<<blocks, BLK, 0, stream>>>(bufAgg, W2, bufH);
    }

    // 7. edge aggregation layer 2 -> out
    {
        long long threads = (long long)N_EDGES * 32;
        int blocks = (int)((threads + BLK - 1) / BLK);
        gcn_edge_agg<<<blocks, BLK, 0, stream>>>(src, dst, dinv, bufH, out, N_EDGES, NCLASS);
    }

    // 8. self-loop + bias + log_softmax (in place on out)
    {
        int blocks = (N_NODES * 32 + BLK - 1) / BLK;
        gcn_post2_logsoftmax<<<blocks, BLK, 0, stream>>>(out, bufH, dinv, b2);
    }
}